// PlaneProjection_24773371364007
// MI455X (gfx1250) — compile-verified
//
#include <hip/hip_runtime.h>
#include <hip/hip_bf16.h>
#include <stdint.h>

typedef float v2f __attribute__((ext_vector_type(2)));
typedef float v8f __attribute__((ext_vector_type(8)));

#define EPSF 1e-12f

// ---------------------------------------------------------------------------
// Kernel 1: per-output-row stats over normal/weight rows.
//   c1 = 1/||w||, c2 = sx/(||n||^2 ||w||), c3 = sx/||w||  with sx = <normal,weight>
// ---------------------------------------------------------------------------
__global__ __launch_bounds__(256) void rowstats_nw_kernel(
    const float* __restrict__ normal, const float* __restrict__ weight,
    float* __restrict__ c1, float* __restrict__ c2, float* __restrict__ c3,
    int IN) {
  __shared__ float s0[256], s1[256], s2[256];
  const int o = blockIdx.x;
  const int t = threadIdx.x;
  const float* nr = normal + (size_t)o * IN;
  const float* wr = weight + (size_t)o * IN;
  float sn = 0.f, sw = 0.f, sx = 0.f;
  for (int i = t; i < IN; i += 256) {
    float nv = nr[i];
    float wv = wr[i];
    sn += nv * nv;
    sw += wv * wv;
    sx += nv * wv;
  }
  s0[t] = sn; s1[t] = sw; s2[t] = sx;
  __syncthreads();
  for (int st = 128; st > 0; st >>= 1) {
    if (t < st) {
      s0[t] += s0[t + st];
      s1[t] += s1[t + st];
      s2[t] += s2[t + st];
    }
    __syncthreads();
  }
  if (t == 0) {
    float inv_n = 1.0f / fmaxf(sqrtf(s0[0]), EPSF);
    float inv_w = 1.0f / fmaxf(sqrtf(s1[0]), EPSF);
    float sxv = s2[0];
    c1[o] = inv_w;
    c2[o] = sxv * inv_n * inv_n * inv_w;
    c3[o] = sxv * inv_w;
  }
}

// ---------------------------------------------------------------------------
// Kernel 2: per-batch-row inverse L2 norm of x.
// ---------------------------------------------------------------------------
__global__ __launch_bounds__(256) void rowstats_x_kernel(
    const float* __restrict__ x, float* __restrict__ invx, int IN) {
  __shared__ float s0[256];
  const int b = blockIdx.x;
  const int t = threadIdx.x;
  const float* xr = x + (size_t)b * IN;
  float s = 0.f;
  for (int i = t; i < IN; i += 256) {
    float v = xr[i];
    s += v * v;
  }
  s0[t] = s;
  __syncthreads();
  for (int st = 128; st > 0; st >>= 1) {
    if (t < st) s0[t] += s0[t + st];
    __syncthreads();
  }
  if (t == 0) invx[b] = 1.0f / fmaxf(sqrtf(s0[0]), EPSF);
}

// ---------------------------------------------------------------------------
// Async global -> LDS staging (CDNA5): 16B per lane, ASYNCcnt-tracked.
// LDS operand is the low 32 bits of the generic shared-space address
// (aperture layout: addr[31:0] == LDS offset). Global operand: 64-bit VGPR.
// ---------------------------------------------------------------------------
__device__ __forceinline__ void async_ld16(const float* g, const float* l) {
  asm volatile("global_load_async_to_lds_b128 %0, %1, off"
               :
               : "v"((uint32_t)(uintptr_t)l), "v"((uint64_t)(uintptr_t)g)
               : "memory");
}

// ---------------------------------------------------------------------------
// Kernel 3: dual GEMM (shared A = x) via V_WMMA_F32_16X16X4_F32 with
// double-buffered async global->LDS staging, fused epilogue:
//   out[b,o] = c1[o]*(x@w^T) - invx[b]*c2[o]*(x@n^T) + c3[o]
//
// Block: 128 threads = 4 waves (2x2 over M,N), block tile 64(M) x 64(N).
// Wave tile: 32x32 = 2x2 WMMA 16x16 tiles, dual f32 accumulators.
// K-step 16, LDS tiles padded to stride 20 floats (conflict-free ds_load_b64,
// 16B-aligned b128 async writes).
// ---------------------------------------------------------------------------
#define LSTR 20  // padded LDS row stride in floats

__global__ __launch_bounds__(128) void gemm_dual_wmma_kernel(
    const float* __restrict__ x, const float* __restrict__ weight,
    const float* __restrict__ normal, const float* __restrict__ c1,
    const float* __restrict__ c2, const float* __restrict__ c3,
    const float* __restrict__ invx, float* __restrict__ out) {
  const int IN = 1024;
  const int OUT = 1024;

  __shared__ __align__(16) float As[2][64 * LSTR];
  __shared__ __align__(16) float Ws[2][64 * LSTR];
  __shared__ __align__(16) float Ns[2][64 * LSTR];

  const int t = threadIdx.x;
  const int lane = t & 31;
  const int wave = t >> 5;
  const int wm = wave >> 1;  // 0..1
  const int wn = wave & 1;   // 0..1

  const int m_blk = blockIdx.y * 64;
  const int n_blk = blockIdx.x * 64;

  const float* xB = x + (size_t)m_blk * IN;
  const float* wB = weight + (size_t)n_blk * IN;
  const float* nB = normal + (size_t)n_blk * IN;

  // Staging map: 64x16 f32 tile = 256 float4; 128 threads -> 2 float4 each.
  const int r0 = t >> 2;             // rows 0..31
  const int cc = (t & 3) << 2;       // float col 0,4,8,12

#define STAGE(BUF, KK)                                                       \
  {                                                                          \
    const int kk_ = (KK);                                                    \
    async_ld16(xB + (size_t)r0 * IN + kk_ + cc, &As[BUF][r0 * LSTR + cc]);   \
    async_ld16(xB + (size_t)(r0 + 32) * IN + kk_ + cc,                       \
               &As[BUF][(r0 + 32) * LSTR + cc]);                             \
    async_ld16(wB + (size_t)r0 * IN + kk_ + cc, &Ws[BUF][r0 * LSTR + cc]);   \
    async_ld16(wB + (size_t)(r0 + 32) * IN + kk_ + cc,                       \
               &Ws[BUF][(r0 + 32) * LSTR + cc]);                             \
    async_ld16(nB + (size_t)r0 * IN + kk_ + cc, &Ns[BUF][r0 * LSTR + cc]);   \
    async_ld16(nB + (size_t)(r0 + 32) * IN + kk_ + cc,                       \
               &Ns[BUF][(r0 + 32) * LSTR + cc]);                             \
  }

  // WMMA fragment addressing (wave32): M/N = lane&15, K offset = (lane>>4)*2.
  const int fr = lane & 15;
  const int koff = (lane >> 4) << 1;
  const int am0 = (wm * 32 + fr) * LSTR;
  const int am1 = (wm * 32 + 16 + fr) * LSTR;
  const int bn0 = (wn * 32 + fr) * LSTR;
  const int bn1 = (wn * 32 + 16 + fr) * LSTR;

  v8f accw00 = {}, accw01 = {}, accw10 = {}, accw11 = {};
  v8f accn00 = {}, accn01 = {}, accn10 = {}, accn11 = {};

  STAGE(0, 0)

  const int KSTEPS = IN / 16;  // 64
  for (int kb = 0; kb < KSTEPS; ++kb) {
    const int cur = kb & 1;
    if (kb + 1 < KSTEPS) {
      STAGE(cur ^ 1, (kb + 1) * 16)
      // 6 prefetch asyncs in flight; wait for current buffer's 6 to land.
      asm volatile("s_wait_asynccnt 0x6" ::: "memory");
    } else {
      asm volatile("s_wait_asynccnt 0x0" ::: "memory");
    }
    __syncthreads();

#pragma unroll
    for (int s = 0; s < 4; ++s) {
      const int kc = s * 4 + koff;
      v2f a0 = *(const v2f*)&As[cur][am0 + kc];
      v2f a1 = *(const v2f*)&As[cur][am1 + kc];
      v2f w0 = *(const v2f*)&Ws[cur][bn0 + kc];
      v2f w1 = *(const v2f*)&Ws[cur][bn1 + kc];
      v2f n0 = *(const v2f*)&Ns[cur][bn0 + kc];
      v2f n1 = *(const v2f*)&Ns[cur][bn1 + kc];

      accw00 = __builtin_amdgcn_wmma_f32_16x16x4_f32(false, a0, false, w0, (short)0, accw00, false, false);
      accw01 = __builtin_amdgcn_wmma_f32_16x16x4_f32(false, a0, false, w1, (short)0, accw01, false, false);
      accw10 = __builtin_amdgcn_wmma_f32_16x16x4_f32(false, a1, false, w0, (short)0, accw10, false, false);
      accw11 = __builtin_amdgcn_wmma_f32_16x16x4_f32(false, a1, false, w1, (short)0, accw11, false, false);
      accn00 = __builtin_amdgcn_wmma_f32_16x16x4_f32(false, a0, false, n0, (short)0, accn00, false, false);
      accn01 = __builtin_amdgcn_wmma_f32_16x16x4_f32(false, a0, false, n1, (short)0, accn01, false, false);
      accn10 = __builtin_amdgcn_wmma_f32_16x16x4_f32(false, a1, false, n0, (short)0, accn10, false, false);
      accn11 = __builtin_amdgcn_wmma_f32_16x16x4_f32(false, a1, false, n1, (short)0, accn11, false, false);
    }
    __syncthreads();  // all waves done reading buf 'cur' before it is restaged
  }
#undef STAGE

  // Epilogue. C/D layout: VGPR r -> M = r + (lane>>4)*8, N = lane&15.
  const int m_base = m_blk + wm * 32;
  const int n_base = n_blk + wn * 32;
  const int nc = lane & 15;
  const int rbase = (lane >> 4) * 8;

#define EPILOG(ACCW, ACCN, MT, NT)                                         \
  {                                                                        \
    const int n = n_base + (NT)*16 + nc;                                   \
    const float C1 = c1[n];                                                \
    const float C2 = c2[n];                                                \
    const float C3 = c3[n];                                                \
    _Pragma("unroll") for (int r = 0; r < 8; ++r) {                        \
      const int b = m_base + (MT)*16 + rbase + r;                          \
      out[(size_t)b * OUT + n] =                                           \
          C1 * (ACCW)[r] - invx[b] * C2 * (ACCN)[r] + C3;                  \
    }                                                                      \
  }

  EPILOG(accw00, accn00, 0, 0)
  EPILOG(accw01, accn01, 0, 1)
  EPILOG(accw10, accn10, 1, 0)
  EPILOG(accw11, accn11, 1, 1)
#undef EPILOG
}

// ---------------------------------------------------------------------------
// Launcher
// ---------------------------------------------------------------------------
extern "C" void kernel_launch(void* const* d_in, const int* in_sizes, int n_in,
                              void* d_out, int out_size, void* d_ws, size_t ws_size,
                              hipStream_t stream) {
  const int B = 512, IN = 1024, OUT = 1024;

  const float* x = (const float*)d_in[0];
  const float* normal = (const float*)d_in[1];
  const float* weight = (const float*)d_in[2];
  float* out = (float*)d_out;

  // Workspace layout (floats): c1[OUT], c2[OUT], c3[OUT], invx[B]
  float* c1 = (float*)d_ws;
  float* c2 = c1 + OUT;
  float* c3 = c2 + OUT;
  float* invx = c3 + OUT;

  rowstats_nw_kernel<<<OUT, 256, 0, stream>>>(normal, weight, c1, c2, c3, IN);
  rowstats_x_kernel<<<B, 256, 0, stream>>>(x, invx, IN);

  dim3 grid(OUT / 64, B / 64);  // (16, 8) = 128 blocks x 4 waves
  gemm_dual_wmma_kernel<<<grid, 128, 0, stream>>>(x, weight, normal, c1, c2, c3,
                                                  invx, out);
}